// Head_66786741452837
// MI455X (gfx1250) — compile-verified
//
#include <hip/hip_runtime.h>
#include <math.h>

#define NEMBD 384
#define HS 64
#define FTHETA 10000.0f

typedef float v2f __attribute__((ext_vector_type(2)));
typedef float v8f __attribute__((ext_vector_type(8)));

__device__ __forceinline__ v8f wmma4(v2f a, v2f b, v8f c) {
  // D(16x16,f32) = A(16x4,f32) * B(4x16,f32) + C
  return __builtin_amdgcn_wmma_f32_16x16x4_f32(
      /*neg_a=*/false, a, /*neg_b=*/false, b,
      /*c_mod=*/(short)0, c, /*reuse_a=*/false, /*reuse_b=*/false);
}

// ---------------------------------------------------------------------------
// Kernel 1: QKV projection + RoPE.  One thread per (row, rotary pair j).
// Block = 256 threads = 8 rows x 32 pairs. x rows staged in LDS.
// ---------------------------------------------------------------------------
__global__ __launch_bounds__(256) void qkv_rope_kernel(
    const float* __restrict__ x, const float* __restrict__ Wq,
    const float* __restrict__ Wk, const float* __restrict__ Wv,
    float* __restrict__ qt, float* __restrict__ kt, float* __restrict__ vo,
    int T)
{
  __shared__ float xs[8 * NEMBD];
  const int row0 = blockIdx.x * 8;
  for (int i = threadIdx.x; i < 8 * NEMBD; i += 256)
    xs[i] = x[(size_t)row0 * NEMBD + i];
  __syncthreads();

  const int local = threadIdx.x >> 5;   // row within block
  const int j     = threadIdx.x & 31;   // rotary pair index 0..31
  const int row   = row0 + local;       // row in [0, B*T)
  const int t     = row % T;            // time index
  const float* xr = &xs[local * NEMBD];

  const float* wq0 = Wq + (size_t)(2 * j)     * NEMBD;
  const float* wq1 = Wq + (size_t)(2 * j + 1) * NEMBD;
  const float* wk0 = Wk + (size_t)(2 * j)     * NEMBD;
  const float* wk1 = Wk + (size_t)(2 * j + 1) * NEMBD;
  const float* wv0 = Wv + (size_t)(2 * j)     * NEMBD;
  const float* wv1 = Wv + (size_t)(2 * j + 1) * NEMBD;

  float q0 = 0.f, q1 = 0.f, k0 = 0.f, k1 = 0.f, v0 = 0.f, v1 = 0.f;
#pragma unroll 4
  for (int c = 0; c < NEMBD; ++c) {
    const float xv = xr[c];
    q0 = fmaf(xv, wq0[c], q0);
    q1 = fmaf(xv, wq1[c], q1);
    k0 = fmaf(xv, wk0[c], k0);
    k1 = fmaf(xv, wk1[c], k1);
    v0 = fmaf(xv, wv0[c], v0);
    v1 = fmaf(xv, wv1[c], v1);
  }

  // RoPE tables exactly as the reference builds them:
  // freq[i] = THETA^(-(2i)/64), i = 0..31
  // emb = [sin(t*freq[0..31]), cos(t*freq[0..31])]
  // cos_tab[j] = emb[2j], sin_tab[j] = emb[2j+1]
  const float lnT = logf(FTHETA);
  const float tf = (float)t;
  float ct, st;
  if (j < 16) {
    const float f0 = expf(-(2.0f * (float)(2 * j))     * (1.0f / 64.0f) * lnT);
    const float f1 = expf(-(2.0f * (float)(2 * j + 1)) * (1.0f / 64.0f) * lnT);
    ct = sinf(tf * f0);
    st = sinf(tf * f1);
  } else {
    const float f0 = expf(-(2.0f * (float)(2 * j - 32)) * (1.0f / 64.0f) * lnT);
    const float f1 = expf(-(2.0f * (float)(2 * j - 31)) * (1.0f / 64.0f) * lnT);
    ct = cosf(tf * f0);
    st = cosf(tf * f1);
  }

  const float qr = q0 * ct + q1 * st;
  const float kr = k0 * ct - k1 * st;

  const size_t base = (size_t)row * HS + 2 * j;
  qt[base]     = qr;
  qt[base + 1] = q1 - qr * st;
  kt[base]     = kr;
  kt[base + 1] = k1 - kr * st;
  vo[base]     = v0;
  vo[base + 1] = v1;
}

// ---------------------------------------------------------------------------
// Kernel 2: flash attention, fp32 WMMA 16x16x4. One wave per 16-query tile.
//
// Fragment layouts (ISA 7.12.2, f32):
//   A(16x4):  lane L -> m = L%16, kloc = 2*(L/16)+v   (v = vgpr 0/1)
//   B(4x16):  lane L -> n = L%16, kloc = 2*(L/16)+v
//   C/D(16x16): vgpr r, lane L -> m = r + 8*(L/16), n = L%16
// ---------------------------------------------------------------------------
__global__ __launch_bounds__(32) void attn_kernel(
    const float* __restrict__ qt, const float* __restrict__ kt,
    const float* __restrict__ vo, float* __restrict__ out, int T)
{
  const int ntiles = T >> 4;
  const int b      = blockIdx.x / ntiles;
  const int qtile  = blockIdx.x % ntiles;
  const int lane   = threadIdx.x;
  const int half   = lane >> 4;
  const int lid    = lane & 15;
  const int qbase  = qtile << 4;
  const size_t rowq = (size_t)b * T + qbase;

  __shared__ float Smem[256];   // 16x16 S / P tile
  __shared__ float Mrow[16];
  __shared__ float Arow[16];
  __shared__ float Lrow[16];

  // Preload Q A-fragments: qA[ks] covers K columns [4ks, 4ks+4)
  v2f qA[16];
  {
    const float* qp = qt + (rowq + lid) * HS + 2 * half;
#pragma unroll
    for (int ks = 0; ks < 16; ++ks)
      qA[ks] = *(const v2f*)(qp + ks * 4);
  }

  v8f o[4];
#pragma unroll
  for (int i = 0; i < 4; ++i) o[i] = v8f{};
  float mrun = -1e30f, lrun = 0.0f;
  const float scale = 0.05103103630798288f;  // 384^-0.5

  const int nkt = qtile + 1;
  for (int ktile = 0; ktile < nkt; ++ktile) {
    const int kbase   = ktile << 4;
    const size_t rowk = (size_t)b * T + kbase;

    // K tile as B-fragments (same per-lane pattern as Q since B(k,n)=K(n,k))
    v2f kB[16];
    {
      const float* kp = kt + (rowk + lid) * HS + 2 * half;
#pragma unroll
      for (int ks = 0; ks < 16; ++ks)
        kB[ks] = *(const v2f*)(kp + ks * 4);
    }

    // S = Q K^T  (16 wmma, K=64 in steps of 4)
    v8f s = v8f{};
#pragma unroll
    for (int ks = 0; ks < 16; ++ks)
      s = wmma4(qA[ks], kB[ks], s);

    // scale + causal mask; stash masked tile in LDS for row reductions
    const int kg = kbase + lid;
#pragma unroll
    for (int r = 0; r < 8; ++r) {
      const int m  = r + 8 * half;
      const int qg = qbase + m;
      const float sv = (kg <= qg) ? s[r] * scale : -1e30f;
      s[r] = sv;
      Smem[m * 16 + lid] = sv;
    }
    __syncthreads();

    // Online-softmax row stats (lanes 0..15, one row each)
    if (lane < 16) {
      float mt = -1e30f;
#pragma unroll
      for (int n = 0; n < 16; ++n) mt = fmaxf(mt, Smem[lane * 16 + n]);
      const float mnew  = fmaxf(mrun, mt);
      const float alpha = __expf(mrun - mnew);
      mrun = mnew;
      lrun *= alpha;
      Mrow[lane] = mnew;
      Arow[lane] = alpha;
    }
    __syncthreads();

    // P = exp(S - m_new) in registers; rescale O; write P to LDS (A reload)
#pragma unroll
    for (int r = 0; r < 8; ++r) {
      const int m = r + 8 * half;
      const float p  = __expf(s[r] - Mrow[m]);
      const float al = Arow[m];
      Smem[m * 16 + lid] = p;
      o[0][r] *= al; o[1][r] *= al; o[2][r] *= al; o[3][r] *= al;
    }
    __syncthreads();

    if (lane < 16) {
      float rs = 0.0f;
#pragma unroll
      for (int n = 0; n < 16; ++n) rs += Smem[lane * 16 + n];
      lrun += rs;
    }

    // O += P V   (P reloaded from LDS in A-fragment layout)
    v2f pA[4];
#pragma unroll
    for (int kk = 0; kk < 4; ++kk)
      pA[kk] = *(const v2f*)(&Smem[lid * 16 + kk * 4 + 2 * half]);
#pragma unroll
    for (int nt = 0; nt < 4; ++nt) {
#pragma unroll
      for (int kk = 0; kk < 4; ++kk) {
        const size_t vrow = rowk + kk * 4 + 2 * half;
        v2f vB;
        vB.x = vo[vrow * HS + nt * 16 + lid];
        vB.y = vo[(vrow + 1) * HS + nt * 16 + lid];
        o[nt] = wmma4(pA[kk], vB, o[nt]);
      }
    }
    __syncthreads();   // protect Smem before next iteration rewrites it
  }

  if (lane < 16) Lrow[lane] = lrun;
  __syncthreads();

#pragma unroll
  for (int r = 0; r < 8; ++r) {
    const int m = r + 8 * half;
    const float inv = 1.0f / Lrow[m];
    float* op = out + (rowq + m) * HS + lid;
    op[0]  = o[0][r] * inv;
    op[16] = o[1][r] * inv;
    op[32] = o[2][r] * inv;
    op[48] = o[3][r] * inv;
  }
}

// ---------------------------------------------------------------------------
extern "C" void kernel_launch(void* const* d_in, const int* in_sizes, int n_in,
                              void* d_out, int out_size, void* d_ws, size_t ws_size,
                              hipStream_t stream) {
  const float* x  = (const float*)d_in[0];
  const float* Wq = (const float*)d_in[1];
  const float* Wk = (const float*)d_in[2];
  const float* Wv = (const float*)d_in[3];

  const int BT = in_sizes[0] / NEMBD;   // B*T = 16384
  const int T  = 4096;
  const int B  = BT / T;

  float* qt = (float*)d_ws;
  float* kt = qt + (size_t)BT * HS;
  float* vv = kt + (size_t)BT * HS;
  float* out = (float*)d_out;

  qkv_rope_kernel<<<BT / 8, 256, 0, stream>>>(x, Wq, Wk, Wv, qt, kt, vv, T);
  attn_kernel<<<B * (T / 16), 32, 0, stream>>>(qt, kt, vv, out, T);
}